// Decoder_88940182765622
// MI455X (gfx1250) — compile-verified
//
#include <hip/hip_runtime.h>
#include <hip/hip_bf16.h>

// Decoder: masked projection + overlap-and-add for MI455X (gfx1250).
//
// Roofline: ~410 MB streamed @ 23.3 TB/s => ~18 us floor; GEMM is only
// 2.14 GFLOP, so we keep exact f32 math with V_WMMA_F32_16X16X4_F32
// (16x16 f32 C tile, K stepped by 4; 128 WMMAs per wave over K=512).
//
// Shapes (fixed by the reference setup):
//   inputs  [16, 4086, 512] f32
//   estmask [16, 4086, 512, 2] f32
//   W       [512, 16] f32
//   out     [16, 2, 32696] f32 ; step = 8, L = 16
//
// Intermediate y is padded to 4096 frame rows so every tile (including the
// 6-row tail tile) stores its full 16x16 C tile unconditionally -> no EXEC
// juggling / branches in the epilogue. The OLA pass only reads rows < 4086.

typedef float v2f __attribute__((ext_vector_type(2)));
typedef float v8f __attribute__((ext_vector_type(8)));

#define BS     16
#define FRAME  4086
#define PFRAME 4096    // padded frame rows in workspace y
#define BASIS  512
#define LOUT   16
#define SPK    2
#define STEP   8
#define OUTLEN 32696   // (FRAME-1)*STEP + LOUT
#define FTILES ((FRAME + 15) / 16)   // 256 (last tile has 6 valid rows)

// ---------------------------------------------------------------------------
// Kernel 1: masked GEMM tile via WMMA f32 16x16x4.
// Block = 64 threads (2 wave32). Wave s handles speaker s.
// LDS: sA[2][16][512] f32 = 64 KB (masked A tiles, zero-padded tail).
// y layout: [BS*SPK][PFRAME][LOUT]
// ---------------------------------------------------------------------------
__global__ __launch_bounds__(64) void masked_proj_wmma(
    const float* __restrict__ inputs,
    const float* __restrict__ estmask,
    const float* __restrict__ W,
    float* __restrict__ y)
{
    __shared__ float sA[SPK][16][BASIS];   // 65536 bytes

    const int tid = threadIdx.x;
    const int f0  = blockIdx.x * 16;
    const int b   = blockIdx.y;
    const int nf  = min(16, FRAME - f0);   // valid frames in this tile

    // ---- Stage: masked = inputs * estmask, both speakers, coalesced float4 ----
    const size_t inBase = ((size_t)b * FRAME + f0) * BASIS;
    const float4* in4 = (const float4*)(inputs + inBase);
    const float4* em4 = (const float4*)(estmask + inBase * 2);

    for (int i = tid; i < 16 * BASIS / 4; i += 64) {   // 2048 float4 units
        const int f  = i >> 7;              // / (BASIS/4)
        const int c4 = (i & 127) << 2;      // starting basis index
        float4 a0, a1;
        if (f < nf) {
            const float4 vin = in4[i];
            const float4 ea  = em4[2 * i + 0];   // (c,s0)(c,s1)(c+1,s0)(c+1,s1)
            const float4 eb  = em4[2 * i + 1];   // (c+2,s0)(c+2,s1)(c+3,s0)(c+3,s1)
            a0 = make_float4(vin.x * ea.x, vin.y * ea.z, vin.z * eb.x, vin.w * eb.z);
            a1 = make_float4(vin.x * ea.y, vin.y * ea.w, vin.z * eb.y, vin.w * eb.w);
        } else {
            a0 = make_float4(0.f, 0.f, 0.f, 0.f);
            a1 = a0;
        }
        *(float4*)&sA[0][f][c4] = a0;
        *(float4*)&sA[1][f][c4] = a1;
    }
    __syncthreads();

    // ---- WMMA phase: wave s computes [16 frames x 16 L], K = 512 ----
    const int s    = tid >> 5;     // wave id == speaker
    const int lane = tid & 31;
    const int half = lane >> 4;    // 0: K pair (k0,k0+1), 1: (k0+2,k0+3)
    const int m    = lane & 15;    // A: row index; B: column index

    // A (16x4 f32): lanes 0-15 hold row m, K = k0..k0+1; lanes 16-31 K = k0+2..k0+3
    const float* arow = &sA[s][m][2 * half];
    // B (4x16 f32): VGPR0 lane<16 -> W[k0][m], lane>=16 -> W[k0+2][m]; VGPR1 +1 row
    const float* wcol = W + (2 * half) * LOUT + m;

    v8f acc = {};
    for (int k0 = 0; k0 < BASIS; k0 += 4) {
        v2f a;
        a[0] = arow[k0];
        a[1] = arow[k0 + 1];
        v2f bm;
        bm[0] = wcol[(k0 + 0) * LOUT];
        bm[1] = wcol[(k0 + 1) * LOUT];
        // (neg_a, A, neg_b, B, c_mod, C, reuse_a, reuse_b)
        acc = __builtin_amdgcn_wmma_f32_16x16x4_f32(
            false, a, false, bm, (short)0, acc, false, false);
    }

    // ---- Store C tile unconditionally (y padded to PFRAME rows) ----
    // C layout: VGPR r -> row r (lanes 0-15) / row r+8 (lanes 16-31), col = m.
    float* yrow = y + ((size_t)(b * SPK + s) * PFRAME + (f0 + 8 * half)) * LOUT + m;
    #pragma unroll
    for (int r = 0; r < 8; ++r) {
        yrow[(size_t)r * LOUT] = acc[r];
    }
}

// ---------------------------------------------------------------------------
// Kernel 2: overlap-and-add, step 8, frame length 16.
// Each output sample has at most two contributors -> deterministic gather
// (no float atomics, bitwise-stable across replays).
// ---------------------------------------------------------------------------
__global__ __launch_bounds__(256) void overlap_add_kernel(
    const float* __restrict__ y,   // [BS*SPK][PFRAME][LOUT]
    float* __restrict__ out)       // [BS*SPK][OUTLEN]
{
    const int g = blockIdx.x * 256 + threadIdx.x;
    if (g >= BS * SPK * OUTLEN) return;

    const int bsi = g / OUTLEN;
    const int p   = g - bsi * OUTLEN;
    const int f1  = p >> 3;        // p / STEP
    const int l   = p & 7;         // p % STEP

    const float* yb = y + (size_t)bsi * PFRAME * LOUT;
    float acc = 0.f;
    if (f1 < FRAME) acc += yb[(size_t)f1 * LOUT + l];
    if (f1 >= 1)    acc += yb[(size_t)(f1 - 1) * LOUT + l + 8];
    out[g] = acc;
}

// ---------------------------------------------------------------------------
extern "C" void kernel_launch(void* const* d_in, const int* in_sizes, int n_in,
                              void* d_out, int out_size, void* d_ws, size_t ws_size,
                              hipStream_t stream) {
    const float* inputs  = (const float*)d_in[0];   // [16,4086,512]
    const float* estmask = (const float*)d_in[1];   // [16,4086,512,2]
    const float* W       = (const float*)d_in[2];   // [512,16]
    // d_in[3] = kernel_size_enc (16), d_in[4] = speech_length (32696): fixed.

    float* y = (float*)d_ws;   // needs 32*4096*16*4 = 8,388,608 bytes

    dim3 grid1(FTILES, BS);
    masked_proj_wmma<<<grid1, 64, 0, stream>>>(inputs, estmask, W, y);

    const int total = BS * SPK * OUTLEN;
    overlap_add_kernel<<<(total + 255) / 256, 256, 0, stream>>>(y, (float*)d_out);
}